// RecurrentGCN_4655744549440
// MI455X (gfx1250) — compile-verified
//
#include <hip/hip_runtime.h>

typedef __attribute__((ext_vector_type(2))) float v2f;
typedef __attribute__((ext_vector_type(8))) float v8f;

#define WAVES_PER_BLOCK 8
// Interleaved weight layout in LDS (floats):
//   base(k, half, col) = k*352 + half*160 + col*2   (pair {W[k0][col], W[k0+1][col]}, k0=4k+2*half)
// half offset 160 == 32 (mod 64) dwords -> the two lane-halves of a b64 fragment
// load hit disjoint bank sets (conflict-free). Per-weight stride 2816 floats.
#define WSTRIDE 2816

__global__ __launch_bounds__(256) void zero_ws_kernel(float* __restrict__ ws, long long count) {
  long long i = (long long)blockIdx.x * blockDim.x + threadIdx.x;
  if (i < count) ws[i] = 0.0f;
}

__global__ __launch_bounds__(256) void degree_kernel(const long long* __restrict__ src,
                                                     float* __restrict__ deg, int nE) {
  int e = blockIdx.x * blockDim.x + threadIdx.x;
  if (e < nE) {
    unsafeAtomicAdd(&deg[src[e]], 1.0f);
  }
}

__global__ __launch_bounds__(256) void rsqrt_kernel(float* __restrict__ deg, int n) {
  int i = blockIdx.x * blockDim.x + threadIdx.x;
  if (i < n) {
    float d = deg[i];
    deg[i] = d > 0.0f ? rsqrtf(d) : 0.0f;
  }
}

// One wave (32 lanes) per edge; lane index = feature. Gather x[src,:] (coalesced
// 128B), scale by norm, scatter-add into tx1[dst,:] with fire-and-forget f32 atomics.
__global__ __launch_bounds__(256) void scatter_kernel(const float* __restrict__ x,
                                                      const long long* __restrict__ src,
                                                      const long long* __restrict__ dst,
                                                      const float* __restrict__ dis,
                                                      float* __restrict__ tx1, int nE) {
  long long gid = (long long)blockIdx.x * blockDim.x + threadIdx.x;
  int e = (int)(gid >> 5);
  int f = (int)(gid & 31);
  if (e < nE) {
    long long s = src[e];
    long long d = dst[e];
    float nrm = -dis[s] * dis[d];
    unsafeAtomicAdd(&tx1[d * 32 + f], nrm * x[s * 32 + f]);
  }
}

// One wave per 16-node tile. fp32 WMMA 16x16x4, K-loop over 32, two N-tiles for
// F_OUT=32. B fragments pre-interleaved in LDS -> one ds_load_b64 per fragment.
// Fuses gates + output projection (butterfly reduce over N halves).
__global__ __launch_bounds__(256) void gru_wmma_kernel(
    const float* __restrict__ x, const float* __restrict__ tx1,
    const float* __restrict__ Wxz0, const float* __restrict__ Wxz1,
    const float* __restrict__ bxz,  const float* __restrict__ bhz,
    const float* __restrict__ Wxh0, const float* __restrict__ Wxh1,
    const float* __restrict__ bxh,  const float* __restrict__ bhh,
    const float* __restrict__ Wl,   const float* __restrict__ bl,
    float* __restrict__ out, int n) {
  __shared__ float sWp[4 * WSTRIDE];   // interleaved B fragments, 4 weight matrices
  __shared__ float sBz[32], sBh[32], sWl[32];
  __shared__ float sbl;

  // Repack weights into the interleaved fragment layout.
  for (int idx = threadIdx.x; idx < 512; idx += 256) {
    int kp   = idx >> 5;        // 0..15, k0 = 2*kp
    int col  = idx & 31;
    int k    = kp >> 1;
    int half = kp & 1;
    int src0 = (2 * kp) * 32 + col;
    int src1 = src0 + 32;
    int base = k * 352 + half * 160 + col * 2;
    sWp[base]                   = Wxz0[src0]; sWp[base + 1]               = Wxz0[src1];
    sWp[WSTRIDE     + base]     = Wxz1[src0]; sWp[WSTRIDE     + base + 1] = Wxz1[src1];
    sWp[2 * WSTRIDE + base]     = Wxh0[src0]; sWp[2 * WSTRIDE + base + 1] = Wxh0[src1];
    sWp[3 * WSTRIDE + base]     = Wxh1[src0]; sWp[3 * WSTRIDE + base + 1] = Wxh1[src1];
  }
  if (threadIdx.x < 32) {
    sBz[threadIdx.x] = bxz[threadIdx.x] + bhz[threadIdx.x];
    sBh[threadIdx.x] = bxh[threadIdx.x] + bhh[threadIdx.x];
    sWl[threadIdx.x] = Wl[threadIdx.x];
  }
  if (threadIdx.x == 0) sbl = bl[0];
  __syncthreads();

  const int wave = threadIdx.x >> 5;
  const int lane = threadIdx.x & 31;
  const int half = lane >> 4;     // selects K pair for A/B fragments
  const int lm   = lane & 15;     // row for A, col for B and C

  long long tile = (long long)blockIdx.x * WAVES_PER_BLOCK + wave;
  long long rowA = tile * 16 + lm;
  long long rowC = rowA < (long long)n ? rowA : (long long)(n - 1);  // clamp tail loads
  const float* xr = x   + rowC * 32;
  const float* tr = tx1 + rowC * 32;

  v8f accZ0 = {}, accZ1 = {}, accH0 = {}, accH1 = {};

#pragma unroll
  for (int k = 0; k < 8; ++k) {
    const int k0  = 4 * k + 2 * half;
    const int off = k * 352 + half * 160 + lm * 2;   // shared base, imm offsets below
    // A fragments (16x4 f32): lane=M, VGPR0/1 = K={2*half, 2*half+1}
    v2f ax; ax.x = xr[k0]; ax.y = xr[k0 + 1];
    v2f at; at.x = tr[k0]; at.y = tr[k0 + 1];
#pragma unroll
    for (int t = 0; t < 2; ++t) {
      const int to = off + t * 32;                   // tile 1: +16 cols = +32 floats
      v2f bz0 = *(const v2f*)&sWp[to];
      v2f bz1 = *(const v2f*)&sWp[to + WSTRIDE];
      v2f bh0 = *(const v2f*)&sWp[to + 2 * WSTRIDE];
      v2f bh1 = *(const v2f*)&sWp[to + 3 * WSTRIDE];
      v8f* aZ = t ? &accZ1 : &accZ0;
      v8f* aH = t ? &accH1 : &accH0;
      *aZ = __builtin_amdgcn_wmma_f32_16x16x4_f32(false, ax, false, bz0, (short)0, *aZ, false, false);
      *aZ = __builtin_amdgcn_wmma_f32_16x16x4_f32(false, at, false, bz1, (short)0, *aZ, false, false);
      *aH = __builtin_amdgcn_wmma_f32_16x16x4_f32(false, ax, false, bh0, (short)0, *aH, false, false);
      *aH = __builtin_amdgcn_wmma_f32_16x16x4_f32(false, at, false, bh1, (short)0, *aH, false, false);
    }
  }

  // C layout: element i of accumulator = (row i + 8*half of tile, feature 16*t + lm)
  float partial[8];
#pragma unroll
  for (int i = 0; i < 8; ++i) partial[i] = 0.0f;

#pragma unroll
  for (int t = 0; t < 2; ++t) {
    const int f = 16 * t + lm;
    const float bz = sBz[f], bh = sBh[f], wl = sWl[f];
    v8f aZ = t ? accZ1 : accZ0;
    v8f aH = t ? accH1 : accH0;
#pragma unroll
    for (int i = 0; i < 8; ++i) {
      float zp = aZ[i] + bz;
      float hp = aH[i] + bh;
      float z  = 1.0f / (1.0f + __expf(-zp));
      float ht = tanhf(hp);
      float h  = (1.0f - z) * ht;          // H = Z*0 + (1-Z)*Ht
      float r  = h > 0.0f ? h : 0.0f;      // relu
      partial[i] += r * wl;
    }
  }

  // Reduce across the 16 lanes of each N-half (features); masks 1..8 stay in-half.
#pragma unroll
  for (int i = 0; i < 8; ++i) {
#pragma unroll
    for (int s = 1; s < 16; s <<= 1)
      partial[i] += __shfl_xor(partial[i], s, 32);
  }

  if (lm == 0) {
    long long base = tile * 16 + 8 * half;
#pragma unroll
    for (int i = 0; i < 8; ++i) {
      long long node = base + i;
      if (node < (long long)n) out[node] = partial[i] + sbl;
    }
  }
}

extern "C" void kernel_launch(void* const* d_in, const int* in_sizes, int n_in,
                              void* d_out, int out_size, void* d_ws, size_t ws_size,
                              hipStream_t stream) {
  // Input order: x, edge_index, then (W{nm}0, W{nm}1, b{nm}) for nm in
  // xz,hz,xr,hr,xh,hh, then Wl, bl. H==0 makes the hz/xr/hr/hh weight matrices
  // dead (only their biases survive).
  const float*     x    = (const float*)d_in[0];
  const long long* ei   = (const long long*)d_in[1];
  const float*     Wxz0 = (const float*)d_in[2];
  const float*     Wxz1 = (const float*)d_in[3];
  const float*     bxz  = (const float*)d_in[4];
  const float*     bhz  = (const float*)d_in[7];
  const float*     Wxh0 = (const float*)d_in[14];
  const float*     Wxh1 = (const float*)d_in[15];
  const float*     bxh  = (const float*)d_in[16];
  const float*     bhh  = (const float*)d_in[19];
  const float*     Wl   = (const float*)d_in[20];
  const float*     bl   = (const float*)d_in[21];
  float* out = (float*)d_out;

  const int n = in_sizes[0] / 32;
  const int E = in_sizes[1] / 2;
  const long long* src = ei;
  const long long* dst = ei + E;

  float* deg = (float*)d_ws;   // n floats; becomes dis in-place
  float* tx1 = deg + n;        // n*32 floats

  long long zc = (long long)n * 33;
  zero_ws_kernel<<<(unsigned)((zc + 255) / 256), 256, 0, stream>>>(deg, zc);
  degree_kernel<<<(E + 255) / 256, 256, 0, stream>>>(src, deg, E);
  rsqrt_kernel<<<(n + 255) / 256, 256, 0, stream>>>(deg, n);

  long long st = (long long)E * 32;
  scatter_kernel<<<(unsigned)((st + 255) / 256), 256, 0, stream>>>(x, src, dst, deg, tx1, E);

  gru_wmma_kernel<<<(n + 127) / 128, 256, 0, stream>>>(
      x, tx1, Wxz0, Wxz1, bxz, bhz, Wxh0, Wxh1, bxh, bhh, Wl, bl, out, n);
}